// NodeGate_60533269070351
// MI455X (gfx1250) — compile-verified
//
#include <hip/hip_runtime.h>
#include <cstdint>

// NodeGate on MI455X: pure HBM-streaming op (AI ~0.34 FLOP/B -> ~22us floor at
// 23.3 TB/s for 512MB of traffic). Per-wave row pipeline built on the CDNA5
// async global->LDS data mover (ASYNCcnt), 4-deep (3 rows prefetched ahead =
// 8KB in flight per wave for latency hiding), wave32 ds_swizzle butterfly
// reductions, nontemporal b128 stores. No WMMA on purpose: the gate MLP is
// rank-1 and a WMMA-based row-sum would destroy load coalescing.

#define F_DIM 512
#define WAVES_PER_BLOCK 8
#define BLOCK_THREADS (WAVES_PER_BLOCK * 32)
#define GRID_BLOCKS 2048
#define TOTAL_WAVES (GRID_BLOCKS * WAVES_PER_BLOCK)   // compile-time: div -> shift
#define NBUF 4                       // pipeline depth (rows in flight)

typedef float v4f __attribute__((ext_vector_type(4)));
typedef int   v4i __attribute__((vector_size(16)));   // matches builtin param type

// ---- CDNA5 async global->LDS copy (one b128 per lane = 512B per wave) ------
__device__ __forceinline__ void async_b128(const float* g, float* l) {
#if __has_builtin(__builtin_amdgcn_global_load_async_to_lds_b128)
  __builtin_amdgcn_global_load_async_to_lds_b128(
      (__attribute__((address_space(1))) v4i*)(const v4i*)(const void*)g,
      (__attribute__((address_space(3))) v4i*)(v4i*)(void*)l,
      /*offset=*/0, /*cpol=*/0);
#else
  asm volatile("global_load_async_to_lds_b128 %0, %1, off"
               :: "v"((uint32_t)(uintptr_t)l), "v"((uint64_t)(uintptr_t)g)
               : "memory");
#endif
}

template <int N>
__device__ __forceinline__ void wait_async() {
  // inline asm so it is also a compiler barrier vs. the LDS reads that follow
  asm volatile("s_wait_asynccnt %0" :: "n"(N) : "memory");
}

// ---- wave32 butterfly sum via ds_swizzle (xor-mask mode) -------------------
template <int M>
__device__ __forceinline__ float swz_xor(float x) {
  return __int_as_float(
      __builtin_amdgcn_ds_swizzle(__float_as_int(x), (M << 10) | 0x1F));
}
__device__ __forceinline__ float wave_sum32(float v) {
  v += swz_xor<16>(v);
  v += swz_xor<8>(v);
  v += swz_xor<4>(v);
  v += swz_xor<2>(v);
  v += swz_xor<1>(v);
  return v;
}

// Issue one 2KB row: 4 async b128 instructions (per-lane addresses)
__device__ __forceinline__ void issue_row(const float* gp, float* lp) {
  async_b128(gp + 0,   lp + 0);
  async_b128(gp + 128, lp + 128);
  async_b128(gp + 256, lp + 256);
  async_b128(gp + 384, lp + 384);
}

__global__ void __launch_bounds__(BLOCK_THREADS)
nodegate_kernel(const float* __restrict__ U, const float* __restrict__ W1,
                const float* __restrict__ b1, const float* __restrict__ W2,
                const float* __restrict__ b2, float* __restrict__ out,
                int nrows) {
  __shared__ float tile[WAVES_PER_BLOCK * NBUF * F_DIM];  // 64 KB / block

  const int lane = threadIdx.x & 31;
  const int wid  = threadIdx.x >> 5;
  const int gw   = blockIdx.x * WAVES_PER_BLOCK + wid;   // global wave id

  float* wbuf = &tile[wid * (NBUF * F_DIM)] + lane * 4;  // per-lane LDS base

  // Gate weights: D=64 -> 2 hidden units per lane (lane, lane+32)
  const float w1a = W1[lane], w1b = W1[lane + 32];
  const float b1a = b1[lane], b1b = b1[lane + 32];
  const float w2a = W2[lane], w2b = W2[lane + 32];
  const float b2v = b2[0];

  // Rows handled by this wave: gw, gw+TOTAL_WAVES, ... (K of them, uniform)
  const int K = (gw < nrows) ? ((nrows - 1 - gw) / TOTAL_WAVES + 1) : 0;
  const size_t stride = (size_t)TOTAL_WAVES * F_DIM;     // row step in floats
  const float* gbase = U   + (size_t)gw * F_DIM + lane * 4;
  float*       obase = out + (size_t)gw * F_DIM + lane * 4;

  // Prologue: prime up to NBUF-1 rows
#pragma unroll
  for (int k = 0; k < NBUF - 1; ++k)
    if (k < K) issue_row(gbase + (size_t)k * stride, wbuf + (k & (NBUF - 1)) * F_DIM);

  for (int k = 0; k < K; ++k) {
    // Keep 3 rows prefetched ahead; wait until this row's 4 loads landed
    // (async loads complete in order, so <=4*ahead outstanding => row ready).
    if (k + 3 < K) {
      issue_row(gbase + 3 * stride, wbuf + ((k + 3) & (NBUF - 1)) * F_DIM);
      wait_async<12>();
    } else if (k + 2 < K) {
      wait_async<8>();
    } else if (k + 1 < K) {
      wait_async<4>();
    } else {
      wait_async<0>();
    }

    // ---- pass 1: row mean (data stays in registers afterwards) ----
    const float* lb = wbuf + (k & (NBUF - 1)) * F_DIM;
    v4f a0 = *(const v4f*)(lb + 0);
    v4f a1 = *(const v4f*)(lb + 128);
    v4f a2 = *(const v4f*)(lb + 256);
    v4f a3 = *(const v4f*)(lb + 384);
    float p = (a0.x + a0.y + a0.z + a0.w) + (a1.x + a1.y + a1.z + a1.w) +
              (a2.x + a2.y + a2.z + a2.w) + (a3.x + a3.y + a3.z + a3.w);
    const float s = wave_sum32(p) * (1.0f / (float)F_DIM);

    // ---- tiny excite MLP: Linear(1,64) -> SiLU -> Linear(64,1) -> sigmoid --
    const float x0 = fmaf(s, w1a, b1a);
    const float x1 = fmaf(s, w1b, b1b);
    const float h0 = x0 / (1.0f + __expf(-x0));   // silu
    const float h1 = x1 / (1.0f + __expf(-x1));
    const float dp = wave_sum32(fmaf(h0, w2a, h1 * w2b));
    const float g  = 1.0f / (1.0f + __expf(-(dp + b2v)));

    // ---- pass 2: scale registers, stream out with NT b128 stores ----
    __builtin_nontemporal_store(a0 * g, (v4f*)(obase + 0));
    __builtin_nontemporal_store(a1 * g, (v4f*)(obase + 128));
    __builtin_nontemporal_store(a2 * g, (v4f*)(obase + 256));
    __builtin_nontemporal_store(a3 * g, (v4f*)(obase + 384));

    gbase += stride;
    obase += stride;
  }
}

extern "C" void kernel_launch(void* const* d_in, const int* in_sizes, int n_in,
                              void* d_out, int out_size, void* d_ws, size_t ws_size,
                              hipStream_t stream) {
  (void)n_in; (void)out_size; (void)d_ws; (void)ws_size;
  const float* U  = (const float*)d_in[0];
  const float* W1 = (const float*)d_in[1];
  const float* b1 = (const float*)d_in[2];
  const float* W2 = (const float*)d_in[3];
  const float* b2 = (const float*)d_in[4];
  float* out = (float*)d_out;
  const int nrows = in_sizes[0] / F_DIM;  // B*N = 131072
  nodegate_kernel<<<GRID_BLOCKS, BLOCK_THREADS, 0, stream>>>(
      U, W1, b1, W2, b2, out, nrows);
}